// LightningIndexer_23570780520555
// MI455X (gfx1250) — compile-verified
//
#include <hip/hip_runtime.h>
#include <math.h>

typedef __attribute__((ext_vector_type(2))) float v2f;
typedef __attribute__((ext_vector_type(8))) float v8f;

#define DIM   768
#define NTOK  4096
#define BATCH 2
#define NH    2
#define KD    8

// workspace layout (floats)
#define QPROJ_OFF 0                                  // B*NH*N*KD = 131072
#define WVALS_OFF (BATCH * NH * NTOK * KD)           // +B*NH*N   =  16384
#define KPROJ_OFF (WVALS_OFF + BATCH * NH * NTOK)    // +B*N*KD   =  65536

// K-reduction over 768 with compile-time B stride so all load offsets are
// immediates. A: lane holds tokens[row][k0+d0 .. +1]; B: lane holds
// W[k0+d0 .. +1][col] (col pre-folded into Bbase).
template <int LDB>
__device__ __forceinline__ v8f proj_k_loop(const float* __restrict__ Ap,
                                           const float* __restrict__ Bbase,
                                           int d0) {
    const float* Bp = Bbase + d0 * LDB;
    const float* A  = Ap + d0;
    v8f acc = {};
#pragma unroll 8
    for (int k0 = 0; k0 < DIM; k0 += 4) {
        const v2f a = *(const v2f*)(A + k0);
        v2f bv;
        bv.x = Bp[k0 * LDB];
        bv.y = Bp[k0 * LDB + LDB];
        acc = __builtin_amdgcn_wmma_f32_16x16x4_f32(false, a, false, bv,
                                                    (short)0, acc, false, false);
    }
    return acc;
}

// ---------------------------------------------------------------------------
// Kernel 1: projections as WMMA GEMM. (B*N x 768) @ (768 x {18 | 8}).
// 512 M-tiles of 16 tokens; 3 wave-jobs per M-tile:
//   job 0: A = query tokens, B = Wq cols 0..15
//   job 1: A = query tokens, B = Wq cols 16..17 (cols >=18 clamped/discarded)
//   job 2: A = key   tokens, B = Wk cols 0..7   (cols >=8  clamped/discarded)
// Grid = 192 blocks x 256 thr (8 waves) = 1536 waves exactly.
// ---------------------------------------------------------------------------
__global__ void li_proj_wmma(const float* __restrict__ qtok,
                             const float* __restrict__ ktok,
                             const float* __restrict__ Wq,
                             const float* __restrict__ Wk,
                             const int* __restrict__ hptr,
                             const int* __restrict__ wptr,
                             float* __restrict__ ws) {
    const int lane  = threadIdx.x & 31;
    const int wave  = threadIdx.x >> 5;
    const int gw    = blockIdx.x * 8 + wave;                       // 0..1535
    const int mtile = __builtin_amdgcn_readfirstlane(gw / 3);      // uniform
    const int job   = __builtin_amdgcn_readfirstlane(gw - mtile * 3);
    const int c16   = lane & 15;
    const int half  = lane >> 4;
    const int d0    = half * 2;

    const int tokbase = mtile * 16;

    v8f acc;
    if (job == 2) {
        const float* Ap = ktok + (size_t)(tokbase + c16) * DIM;
        acc = proj_k_loop<KD>(Ap, Wk + (c16 < 8 ? c16 : 7), d0);
    } else {
        const float* Ap = qtok + (size_t)(tokbase + c16) * DIM;
        const int colofs = (job == 1) ? (16 + (c16 < 2 ? c16 : 1)) : c16;
        acc = proj_k_loop<18>(Ap, Wq + colofs, d0);
    }

    float* qproj = ws + QPROJ_OFF;
    float* wvals = ws + WVALS_OFF;
    float* kproj = ws + KPROJ_OFF;

    // M-tiles never straddle a batch (NTOK % 16 == 0): b uniform per wave.
    const int b  = tokbase / NTOK;
    const int n0 = (tokbase - b * NTOK) + half * 8;  // first row this lane owns

    if (job <= 1) {
        const int col = (job == 1) ? 16 + c16 : c16;
        if (col < 18) {
            const int h = col / 9, r = col % 9;
            float* dst;
            int    stride;
            float  mult;
            if (r < KD) {
                dst    = qproj + (((size_t)b * NH + h) * NTOK + n0) * KD + r;
                stride = KD;
                mult   = 1.0f;
            } else {
                // fold scale = key_dim^-0.5 into w (relu(s*x)=s*relu(x), s>0)
                dst    = wvals + ((size_t)b * NH + h) * NTOK + n0;
                stride = 1;
                mult   = 0.35355339059327373f;
            }
#pragma unroll
            for (int v = 0; v < 8; ++v) dst[v * stride] = acc[v] * mult;
        }
    } else if (c16 < KD) {
        const int   height = hptr[0], width = wptr[0];
        const int   d      = c16;
        const bool  usey   = d < 4;
        const bool  iscos  = (d & 1) != 0;
        // d_half = 4 -> freq = [1, 10000^-0.5] = [1, 0.01]
        const float freq   = (((d & 3) >> 1) == 0) ? 1.0f : 0.01f;
        int hm1 = height - 1; if (hm1 < 1) hm1 = 1;
        int wm1 = width  - 1; if (wm1 < 1) wm1 = 1;
        const float denom = usey ? (float)hm1 : (float)wm1;
        const float fs    = freq * (usey ? (float)height : (float)width);
        int yi = n0 / width;              // single integer division per lane
        int xi = n0 - yi * width;
        float* dst = kproj + ((size_t)b * NTOK + n0) * KD + d;
        for (int v = 0; v < 8; ++v) {
            const float idx = (float)(usey ? yi : xi);
            const float ang = (idx / denom) * fs;
            const float pe  = iscos ? cosf(ang) : sinf(ang);
            dst[v * KD] = acc[v] + pe;
            if (++xi == width) { xi = 0; ++yi; }
        }
    }
}

// ---------------------------------------------------------------------------
// Kernel 2: scores via V_WMMA_F32_16X16X4_F32 (K=8 as two chained WMMAs),
// per-head relu (single v_max_num via ternary), weighted combine, bias, store.
// grid = (B * N/16, 8), block = 256 (8 waves); each wave: 4 k-tiles of 16.
// ---------------------------------------------------------------------------
__global__ void li_main(const float* __restrict__ bias_ptr,
                        const float* __restrict__ ws,
                        float* __restrict__ out) {
    const float* qproj = ws + QPROJ_OFF;
    const float* wvals = ws + WVALS_OFF;
    const float* kproj = ws + KPROJ_OFF;

    const int lane  = threadIdx.x & 31;
    const int wave  = threadIdx.x >> 5;
    const int b     = blockIdx.x >> 8;       // / (N/16)
    const int qt    = blockIdx.x & 255;
    const int baseq = qt * 16;
    const int c16   = lane & 15;
    const int half  = lane >> 4;             // 0: K=0,1(+4) ; 1: K=2,3(+4)
    const int d0    = half * 2;

    const float bias = bias_ptr[0];

    // A operands (16x4 f32 layout): lane holds q[baseq + lane%16][d0 .. d0+1]
    const int rowA = baseq + c16;
    const float* p0 = qproj + (((size_t)b * NH + 0) * NTOK + rowA) * KD;
    const float* p1 = qproj + (((size_t)b * NH + 1) * NTOK + rowA) * KD;
    const v2f a0_lo = *(const v2f*)(p0 + d0);
    const v2f a0_hi = *(const v2f*)(p0 + 4 + d0);
    const v2f a1_lo = *(const v2f*)(p1 + d0);
    const v2f a1_hi = *(const v2f*)(p1 + 4 + d0);

    // per-row weights for this lane's C/D rows: M = v + 8*half
    float w0[8], w1[8];
    {
        const float* pw0 = wvals + ((size_t)b * NH + 0) * NTOK + baseq + half * 8;
        const float* pw1 = wvals + ((size_t)b * NH + 1) * NTOK + baseq + half * 8;
#pragma unroll
        for (int v = 0; v < 8; ++v) { w0[v] = pw0[v]; w1[v] = pw1[v]; }
    }

    const int kbase0 = blockIdx.y * 512 + wave * 64;
#pragma unroll
    for (int t = 0; t < 4; ++t) {
        const int kbase = kbase0 + t * 16;
        const int colB  = kbase + c16;
        // B operand (4x16 f32 layout): lane holds k[kbase + lane%16][d0 .. d0+1]
        const float* pk = kproj + ((size_t)b * NTOK + colB) * KD;
        const v2f b_lo = *(const v2f*)(pk + d0);
        const v2f b_hi = *(const v2f*)(pk + 4 + d0);

        v8f acc0 = {};
        v8f acc1 = {};
        acc0 = __builtin_amdgcn_wmma_f32_16x16x4_f32(false, a0_lo, false, b_lo,
                                                     (short)0, acc0, false, false);
        acc0 = __builtin_amdgcn_wmma_f32_16x16x4_f32(false, a0_hi, false, b_hi,
                                                     (short)0, acc0, false, false);
        acc1 = __builtin_amdgcn_wmma_f32_16x16x4_f32(false, a1_lo, false, b_lo,
                                                     (short)0, acc1, false, false);
        acc1 = __builtin_amdgcn_wmma_f32_16x16x4_f32(false, a1_hi, false, b_hi,
                                                     (short)0, acc1, false, false);

        // C/D layout: VGPR v -> row (v + 8*half), col = lane%16
        const size_t outb = ((size_t)b * NTOK + baseq + half * 8) * NTOK + colB;
#pragma unroll
        for (int v = 0; v < 8; ++v) {
            const float x0 = acc0[v];
            const float x1 = acc1[v];
            const float r0 = x0 > 0.0f ? x0 : 0.0f;   // single v_max_num_f32
            const float r1 = x1 > 0.0f ? x1 : 0.0f;
            out[outb + (size_t)v * NTOK] =
                fmaf(w0[v], r0, fmaf(w1[v], r1, bias));
        }
    }
}

extern "C" void kernel_launch(void* const* d_in, const int* in_sizes, int n_in,
                              void* d_out, int out_size, void* d_ws, size_t ws_size,
                              hipStream_t stream) {
    (void)in_sizes; (void)n_in; (void)out_size; (void)ws_size;
    const float* qtok = (const float*)d_in[0];
    const float* ktok = (const float*)d_in[1];
    const float* Wq   = (const float*)d_in[2];
    const float* Wk   = (const float*)d_in[3];
    const float* bias = (const float*)d_in[4];
    const int*   h    = (const int*)d_in[5];
    const int*   w    = (const int*)d_in[6];
    float* ws  = (float*)d_ws;
    float* out = (float*)d_out;

    // 512 M-tiles * 3 jobs = 1536 waves = 192 blocks * 8 waves (exact)
    li_proj_wmma<<<dim3(192), 256, 0, stream>>>(qtok, ktok, Wq, Wk, h, w, ws);
    li_main<<<dim3(BATCH * (NTOK / 16), 8), 256, 0, stream>>>(bias, ws, out);
}